// GraphSAGE_7327214207545
// MI455X (gfx1250) — compile-verified
//
#include <hip/hip_runtime.h>

#define HIDDEN 128
#define NUM_LAYERS 3
#define ROWS_PER_BLOCK 80      // 50000 = 625 * 80, exact
#define ROW_TILES 5            // 80 / 16
#define LDS_STRIDE 132         // 128 + 4 pad: bank = (4m + kk) % 64 -> conflict-free

typedef __attribute__((ext_vector_type(2))) float v2f;
typedef __attribute__((ext_vector_type(8))) float v8f;

// ---------------------------------------------------------------------------
// Degree: one thread per edge, fp32 hardware atomic add on deg[dst].
// ---------------------------------------------------------------------------
__global__ void sage_deg_kernel(const int* __restrict__ dst,
                                float* __restrict__ deg, int nEdges) {
    int e = blockIdx.x * blockDim.x + threadIdx.x;
    if (e < nEdges) {
        unsafeAtomicAdd(&deg[dst[e]], 1.0f);
    }
}

__global__ void sage_invdeg_kernel(float* __restrict__ deg, int n) {
    int i = blockIdx.x * blockDim.x + threadIdx.x;
    if (i < n) {
        deg[i] = 1.0f / fmaxf(deg[i], 1.0f);
    }
}

// ---------------------------------------------------------------------------
// Scatter-add: one wave32 per edge. Each lane gathers a float4 (16B) of the
// source row (coalesced 512B per row) and issues 4 global_atomic_add_f32
// into the destination row. h and agg are L2-resident (192MB L2).
// ---------------------------------------------------------------------------
__global__ void sage_scatter_kernel(const float* __restrict__ h,
                                    const int* __restrict__ src,
                                    const int* __restrict__ dst,
                                    float* __restrict__ agg, int nEdges) {
    int gid  = blockIdx.x * blockDim.x + threadIdx.x;
    int e    = gid >> 5;
    int lane = gid & 31;
    if (e >= nEdges) return;
    int s = src[e];
    int d = dst[e];
    const float4 v = ((const float4*)(h + (size_t)s * HIDDEN))[lane];
    float* a = agg + (size_t)d * HIDDEN + lane * 4;
    unsafeAtomicAdd(a + 0, v.x);
    unsafeAtomicAdd(a + 1, v.y);
    unsafeAtomicAdd(a + 2, v.z);
    unsafeAtomicAdd(a + 3, v.w);
}

// ---------------------------------------------------------------------------
// Fused SAGE layer GEMM on the f32 WMMA pipe, LDS-staged A operands:
//   out[80 rows x 128 cols per block] =
//       (agg*inv_deg) @ Wl^T + h @ Wr^T + bl   (+ ReLU)
//
// Block = 256 threads = 8 wave32. Block stages its 80 rows of agg (pre-scaled
// by inv_deg) and h into LDS once; wave w owns columns [16w, 16w+16) and runs
// 5 row tiles, so each B fragment (weights, global/L2) feeds 5 WMMAs.
//
// Fragment layouts (ISA 7.12.2, 32-bit 16x4 A / 4x16 B / 16x16 C):
//   A: lane -> row M = lane&15, holds K = kk, kk+1   (kk = k + 2*(lane>=16))
//   B: lane -> col N = lane&15, holds K = kk, kk+1;  B[k][n] = W[n][k]
//   C/D: vgpr v, lane -> (M = v + 8*(lane>=16), N = lane&15)
//
// In-place (out == h) is safe: all global reads of h happen at staging,
// before __syncthreads(); stores happen after the K loop, and blocks own
// disjoint row ranges.
// ---------------------------------------------------------------------------
__global__ __launch_bounds__(256) void sage_gemm_kernel(
        const float* __restrict__ agg, const float* __restrict__ h,
        const float* __restrict__ Wl,  const float* __restrict__ bl,
        const float* __restrict__ Wr,  const float* __restrict__ invdeg,
        float* __restrict__ out, int relu) {
    __shared__ float aggS[ROWS_PER_BLOCK * LDS_STRIDE];
    __shared__ float hS[ROWS_PER_BLOCK * LDS_STRIDE];

    const int tid     = threadIdx.x;
    const int rowBase = blockIdx.x * ROWS_PER_BLOCK;

    // ---- Stage A operands: 80x128 floats per matrix = 2560 float4, 10/thread.
    // agg is scaled by inv_deg here, so the K-loop is pure loads + WMMA.
#pragma unroll
    for (int i = 0; i < 10; ++i) {
        const int idx = tid + i * 256;     // float4 index
        const int row = idx >> 5;          // 32 float4 per row
        const int c4  = idx & 31;
        const size_t goff = (size_t)(rowBase + row) * HIDDEN;
        const float inv = invdeg[rowBase + row];
        float4 a  = ((const float4*)(agg + goff))[c4];
        float4 hv = ((const float4*)(h   + goff))[c4];
        a.x *= inv; a.y *= inv; a.z *= inv; a.w *= inv;
        *(float4*)&aggS[row * LDS_STRIDE + c4 * 4] = a;
        *(float4*)&hS[row * LDS_STRIDE + c4 * 4]   = hv;
    }
    __syncthreads();

    const int lane   = tid & 31;
    const int wave   = tid >> 5;
    const int li     = lane & 15;
    const int hiHalf = lane >> 4;          // 0 or 1
    const int koff   = hiHalf << 1;        // 0 or 2
    const int n      = wave * 16 + li;     // B column this lane feeds

    const float* wlRow = Wl + (size_t)n * HIDDEN;
    const float* wrRow = Wr + (size_t)n * HIDDEN;

    v8f acc[ROW_TILES];
#pragma unroll
    for (int t = 0; t < ROW_TILES; ++t)
        acc[t] = (v8f){0.f, 0.f, 0.f, 0.f, 0.f, 0.f, 0.f, 0.f};

#pragma unroll 4
    for (int k = 0; k < HIDDEN; k += 4) {
        const int kk = k + koff;           // even -> 8B-aligned loads
        v2f bL = *(const v2f*)(wlRow + kk);
        v2f bR = *(const v2f*)(wrRow + kk);
#pragma unroll
        for (int t = 0; t < ROW_TILES; ++t) {
            const int srow = t * 16 + li;
            v2f aA = *(const v2f*)&aggS[srow * LDS_STRIDE + kk];
            v2f aH = *(const v2f*)&hS[srow * LDS_STRIDE + kk];
            acc[t] = __builtin_amdgcn_wmma_f32_16x16x4_f32(
                         false, aA, false, bL, (short)0, acc[t], false, false);
            acc[t] = __builtin_amdgcn_wmma_f32_16x16x4_f32(
                         false, aH, false, bR, (short)0, acc[t], false, false);
        }
    }

    const float bias = bl[n];
#pragma unroll
    for (int t = 0; t < ROW_TILES; ++t) {
        const int mrow = rowBase + t * 16 + (hiHalf << 3);
#pragma unroll
        for (int v = 0; v < 8; ++v) {
            float val = acc[t][v] + bias;
            if (relu) val = fmaxf(val, 0.0f);
            out[(size_t)(mrow + v) * HIDDEN + wave * 16 + li] = val;
        }
    }
}

// ---------------------------------------------------------------------------
// Driver: deg -> inv_deg once; per layer memset agg, scatter, fused WMMA GEMM.
// Workspace: [deg: N floats][agg: N*128][hbuf: N*128]  (~51.4 MB)
// Layer 0: x -> hbuf, layer 1: hbuf -> hbuf (in-place), layer 2: hbuf -> out.
// ---------------------------------------------------------------------------
extern "C" void kernel_launch(void* const* d_in, const int* in_sizes, int n_in,
                              void* d_out, int out_size, void* d_ws, size_t ws_size,
                              hipStream_t stream) {
    const float* x   = (const float*)d_in[0];
    const int*   ei  = (const int*)d_in[1];
    const float* Wl  = (const float*)d_in[2];
    const float* bl  = (const float*)d_in[3];
    const float* Wr  = (const float*)d_in[4];

    const int nNodes = in_sizes[0] / HIDDEN;   // 50000
    const int nEdges = in_sizes[1] / 2;        // 600000
    const int* src = ei;
    const int* dst = ei + nEdges;

    float* deg  = (float*)d_ws;
    float* agg  = deg + nNodes;
    float* hbuf = agg + (size_t)nNodes * HIDDEN;

    // Degrees and reciprocal (computed once, reused across layers).
    hipMemsetAsync(deg, 0, (size_t)nNodes * sizeof(float), stream);
    sage_deg_kernel<<<(nEdges + 255) / 256, 256, 0, stream>>>(dst, deg, nEdges);
    sage_invdeg_kernel<<<(nNodes + 255) / 256, 256, 0, stream>>>(deg, nNodes);

    const float* hin = x;
    for (int layer = 0; layer < NUM_LAYERS; ++layer) {
        hipMemsetAsync(agg, 0, (size_t)nNodes * HIDDEN * sizeof(float), stream);

        long long scatterThreads = (long long)nEdges * 32;
        sage_scatter_kernel<<<(unsigned)((scatterThreads + 255) / 256), 256, 0, stream>>>(
            hin, src, dst, agg, nEdges);

        float* outp = (layer == NUM_LAYERS - 1) ? (float*)d_out : hbuf;
        sage_gemm_kernel<<<nNodes / ROWS_PER_BLOCK, 256, 0, stream>>>(
            agg, hin,
            Wl + (size_t)layer * HIDDEN * HIDDEN,
            bl + (size_t)layer * HIDDEN,
            Wr + (size_t)layer * HIDDEN * HIDDEN,
            deg, outp, (layer < NUM_LAYERS - 1) ? 1 : 0);

        hin = outp;
    }
}